// GaussianRecurrent_60739427500105
// MI455X (gfx1250) — compile-verified
//
#include <hip/hip_runtime.h>

// ---------------------------------------------------------------------------
// GaussianRecurrent log-likelihood, MI455X (gfx1250).
//
// Closed-form reduction of the scan (coefficients telescope):
//   den_t  = pv + (t-1)*pc
//   mu_t   = (pc*S_t + (pv-pc)*mu0) / den_t,   S_t = exclusive prefix sum of z
//   var_t  = (pv-pc)*(den_t+pc)/den_t
//   sum_t -0.5*log(2*pi*var_t) telescopes to a per-dim constant L0_d.
//   data term per element: u = den*z - pc*S - m0;  -u^2 / (2(pv-pc)*den*(den+pc))
//
// Memory-bound: 128 MB of z @ 23.3 TB/s ~= 5.5 us floor. 128 MB < 192 MB L2,
// so the two-pass scheme (pass1 sums fill L2, pass3 re-reads from L2) costs
// roughly one HBM pass. No matrix contraction exists -> WMMA not applicable;
// CDNA5 paths used: deep async global->LDS pipeline (ASYNCcnt), prefetch.
// ---------------------------------------------------------------------------

#define DIMS   512
#define TPB    128   // 4 wave32 per block; each thread owns 4 columns (b128)
#define NG     32    // scan groups
#define NSTAGE 16    // staging ring buffers in LDS (32 KB)
#define DEPTH  8     // async copies in flight per wave

__device__ __forceinline__ float4 ld4(const float* p) { return *(const float4*)p; }

// -------- Pass 1: per time-chunk column sums (deep async global->LDS) -------
__global__ void gr_pass1(const float* __restrict__ z, float* __restrict__ bsum,
                         int T, int chunk)
{
  const int b = blockIdx.x, tid = threadIdx.x;
  const int d0 = tid * 4;
  const int t0 = b * chunk;
  const int tend = (t0 + chunk < T) ? (t0 + chunk) : T;
  const int nrows = tend - t0;
  float4 s = make_float4(0.f, 0.f, 0.f, 0.f);

  __shared__ __align__(16) float stage[NSTAGE][DIMS];

  const int pro = (nrows < DEPTH) ? nrows : DEPTH;
  for (int j = 0; j < pro; ++j) {
    unsigned lptr = (unsigned)(size_t)&stage[j][d0];
    const float* gptr = z + (size_t)(t0 + j) * DIMS + d0;
    asm volatile("global_load_async_to_lds_b128 %0, %1, off"
                 :: "v"(lptr), "v"(gptr) : "memory");
  }
  for (int i = 0; i < nrows; ++i) {
    if (i + DEPTH < nrows) {
      // copies i+1..i+DEPTH-1 may remain outstanding; copy i must be done
      asm volatile("s_wait_asynccnt 7" ::: "memory");
    } else {
      asm volatile("s_wait_asynccnt 0" ::: "memory");
    }
    // Each lane reads back only the 16B it copied itself: no barrier needed.
    const float4 zv = *(const float4*)&stage[i & (NSTAGE - 1)][d0];
    s.x += zv.x; s.y += zv.y; s.z += zv.z; s.w += zv.w;
    if (i + DEPTH < nrows) {
      unsigned lptr = (unsigned)(size_t)&stage[(i + DEPTH) & (NSTAGE - 1)][d0];
      const float* gptr = z + (size_t)(t0 + i + DEPTH) * DIMS + d0;
      asm volatile("global_load_async_to_lds_b128 %0, %1, off"
                   :: "v"(lptr), "v"(gptr) : "memory");
    }
  }
  *(float4*)(bsum + (size_t)b * DIMS + d0) = s;
}

// -------- Pass 2a: group sums over G chunks ---------------------------------
__global__ void gr_pass2a(const float* __restrict__ bsum, float* __restrict__ gsum,
                          int G)
{
  const int g = blockIdx.x, d = threadIdx.x;
  float s = 0.f;
  const float* p = bsum + (size_t)g * G * DIMS + d;
  for (int j = 0; j < G; ++j) s += p[(size_t)j * DIMS];
  gsum[(size_t)g * DIMS + d] = s;
}

// -------- Pass 2b: per-dim constants + group-level exclusive scan + L0 ------
__global__ void gr_pass2b(const float* __restrict__ var_vbl,
                          const float* __restrict__ corr_vbl,
                          const float* __restrict__ prior_mu,
                          float* __restrict__ gsum,
                          float* __restrict__ pv, float* __restrict__ pc,
                          float* __restrict__ m0, double* __restrict__ L0, int T)
{
  const int d = threadIdx.x;
  const float vv = var_vbl[d];
  const float sp = fmaxf(vv, 0.f) + log1pf(expf(-fabsf(vv)));   // softplus
  const float PV = sp * sp;
  const float cg = corr_vbl[d];
  const float SG = 1.f / (1.f + expf(-cg));                     // sigmoid
  const float PC = SG * PV;
  pv[d] = PV; pc[d] = PC; m0[d] = (PV - PC) * prior_mu[d];

  // Telescoped sum of the -0.5*log(2*pi*var_t) terms over all t, for dim d.
  const double l0 = -0.5 * ((double)T * 1.8378770664093454  /* log(2*pi) */
                            + (double)(T - 1) * log((double)(PV - PC))
                            + log((double)PV + (double)(T - 1) * (double)PC));

  // Exclusive scan over the NG group sums (in place).
  float run = 0.f;
  for (int g = 0; g < NG; ++g) {
    const float tmp = gsum[(size_t)g * DIMS + d];
    gsum[(size_t)g * DIMS + d] = run;
    run += tmp;
  }

  __shared__ double red[DIMS];
  red[d] = l0;
  __syncthreads();
  for (int k = DIMS / 2; k > 0; k >>= 1) {
    if (d < k) red[d] += red[d + k];
    __syncthreads();
  }
  if (d == 0) *L0 = red[0];
}

// -------- Pass 2c: exclusive scan of chunk sums within each group (in place)
__global__ void gr_pass2c(float* __restrict__ bsum, const float* __restrict__ gsum,
                          int G)
{
  const int g = blockIdx.x, d = threadIdx.x;
  float run = gsum[(size_t)g * DIMS + d];
  float* p = bsum + (size_t)g * G * DIMS + d;
  for (int j = 0; j < G; ++j) {
    const float tmp = p[(size_t)j * DIMS];
    p[(size_t)j * DIMS] = run;
    run += tmp;
  }
}

// -------- Pass 3: main streaming pass (reads z from L2) ---------------------
__global__ void gr_pass3(const float* __restrict__ z, const float* __restrict__ boffs,
                         const float* __restrict__ pv, const float* __restrict__ pc,
                         const float* __restrict__ m0, double* __restrict__ partial,
                         int T, int chunk)
{
  const int b = blockIdx.x, tid = threadIdx.x;
  const int d0 = tid * 4;
  const int t0 = b * chunk;
  const int tend = (t0 + chunk < T) ? (t0 + chunk) : T;

  const float4 PV = ld4(pv + d0);
  const float4 PC = ld4(pc + d0);
  const float4 M0 = ld4(m0 + d0);
  float4 S = ld4(boffs + (size_t)b * DIMS + d0);   // exclusive prefix at t0

  float4 den;
  den.x = fmaf((float)(t0 - 1), PC.x, PV.x);
  den.y = fmaf((float)(t0 - 1), PC.y, PV.y);
  den.z = fmaf((float)(t0 - 1), PC.z, PV.z);
  den.w = fmaf((float)(t0 - 1), PC.w, PV.w);
  float4 K2;
  K2.x = 2.f * (PV.x - PC.x);
  K2.y = 2.f * (PV.y - PC.y);
  K2.z = 2.f * (PV.z - PC.z);
  K2.w = 2.f * (PV.w - PC.w);

  // Four independent f32 accumulators: no f64 chain in the hot loop.
  float ax = 0.f, ay = 0.f, az = 0.f, aw = 0.f;
  for (int t = t0; t < tend; ++t) {
    const float* row = z + (size_t)t * DIMS + d0;
    if (t + 8 < tend) __builtin_prefetch(row + 8 * DIMS, 0, 0);  // global_prefetch_b8
    const float4 zv = ld4(row);
    {
      const float q = fmaf(PC.x, S.x, M0.x);
      const float u = fmaf(den.x, zv.x, -q);
      const float w = K2.x * (den.x * (den.x + PC.x));
      ax = fmaf(u * u, __builtin_amdgcn_rcpf(w), ax);
      S.x += zv.x; den.x += PC.x;
    }
    {
      const float q = fmaf(PC.y, S.y, M0.y);
      const float u = fmaf(den.y, zv.y, -q);
      const float w = K2.y * (den.y * (den.y + PC.y));
      ay = fmaf(u * u, __builtin_amdgcn_rcpf(w), ay);
      S.y += zv.y; den.y += PC.y;
    }
    {
      const float q = fmaf(PC.z, S.z, M0.z);
      const float u = fmaf(den.z, zv.z, -q);
      const float w = K2.z * (den.z * (den.z + PC.z));
      az = fmaf(u * u, __builtin_amdgcn_rcpf(w), az);
      S.z += zv.z; den.z += PC.z;
    }
    {
      const float q = fmaf(PC.w, S.w, M0.w);
      const float u = fmaf(den.w, zv.w, -q);
      const float w = K2.w * (den.w * (den.w + PC.w));
      aw = fmaf(u * u, __builtin_amdgcn_rcpf(w), aw);
      S.w += zv.w; den.w += PC.w;
    }
  }
  const double acc = -((double)ax + (double)ay + (double)az + (double)aw);

  __shared__ double red[TPB];
  red[tid] = acc;
  __syncthreads();
  for (int k = TPB / 2; k > 0; k >>= 1) {
    if (tid < k) red[tid] += red[tid + k];
    __syncthreads();
  }
  if (tid == 0) partial[b] = red[0];
}

// -------- Pass 4: fixed-order final reduction -------------------------------
__global__ void gr_pass4(const double* __restrict__ partial,
                         const double* __restrict__ L0,
                         float* __restrict__ out, int nb)
{
  __shared__ double red[256];
  const int tid = threadIdx.x;
  double s = 0.0;
  for (int i = tid; i < nb; i += 256) s += partial[i];
  red[tid] = s;
  __syncthreads();
  for (int k = 128; k > 0; k >>= 1) {
    if (tid < k) red[tid] += red[tid + k];
    __syncthreads();
  }
  if (tid == 0) out[0] = (float)(red[0] + *L0);
}

// ---------------------------------------------------------------------------
extern "C" void kernel_launch(void* const* d_in, const int* in_sizes, int n_in,
                              void* d_out, int out_size, void* d_ws, size_t ws_size,
                              hipStream_t stream)
{
  const float* z   = (const float*)d_in[0];
  const float* vv  = (const float*)d_in[1];
  const float* cv  = (const float*)d_in[2];
  const float* pmu = (const float*)d_in[3];
  const int D = DIMS;                       // reference DIM = 512
  const int T = in_sizes[0] / D;

  auto need = [&](int nb) -> size_t {
    return (size_t)nb * D * 4 + (size_t)NG * D * 4 + 3u * D * 4 + 8 + (size_t)nb * 8 + 64;
  };
  int NB = 1024;
  if (ws_size < need(1024)) NB = 256;
  if (ws_size < need(256))  NB = 32;
  const int G = NB / NG;
  const int chunk = (T + NB - 1) / NB;

  char* w = (char*)d_ws;
  float* bsum = (float*)w;            w += (size_t)NB * D * 4;
  float* gsum = (float*)w;            w += (size_t)NG * D * 4;
  float* pvb  = (float*)w;            w += (size_t)D * 4;
  float* pcb  = (float*)w;            w += (size_t)D * 4;
  float* m0b  = (float*)w;            w += (size_t)D * 4;
  double* L0  = (double*)w;           w += 8;
  double* partial = (double*)w;

  gr_pass1 <<<NB, TPB, 0, stream>>>(z, bsum, T, chunk);
  gr_pass2a<<<NG, D,   0, stream>>>(bsum, gsum, G);
  gr_pass2b<<<1,  D,   0, stream>>>(vv, cv, pmu, gsum, pvb, pcb, m0b, L0, T);
  gr_pass2c<<<NG, D,   0, stream>>>(bsum, gsum, G);
  gr_pass3 <<<NB, TPB, 0, stream>>>(z, bsum, pvb, pcb, m0b, partial, T, chunk);
  gr_pass4 <<<1,  256, 0, stream>>>(partial, L0, (float*)d_out, NB);
}